// RVSoftmax_42855183679902
// MI455X (gfx1250) — compile-verified
//
#include <hip/hip_runtime.h>
#include <hip/hip_bf16.h>

typedef float v2f  __attribute__((ext_vector_type(2)));
typedef float v4f  __attribute__((ext_vector_type(4)));
typedef float v8f  __attribute__((ext_vector_type(8)));

#define BATCH 256
#define FDIM  512

// ---------------------------------------------------------------------------
// Kernel 1: per-row softmax. One wave (32 lanes) per batch row, 16 elems/lane,
// pure cross-lane reductions (wave32 shuffles).
// ---------------------------------------------------------------------------
__global__ void rvs_softmax_kernel(const float* __restrict__ mu,
                                   float* __restrict__ p) {
    const int b    = blockIdx.x;
    const int lane = threadIdx.x;            // 0..31
    const float* row = mu + (size_t)b * FDIM;

    float v[16];
    float m = -INFINITY;
    #pragma unroll
    for (int j = 0; j < 16; ++j) {
        v[j] = row[lane + 32 * j];
        m = fmaxf(m, v[j]);
    }
    #pragma unroll
    for (int off = 16; off > 0; off >>= 1)
        m = fmaxf(m, __shfl_xor(m, off, 32));

    float sum = 0.0f;
    #pragma unroll
    for (int j = 0; j < 16; ++j) {
        v[j] = __expf(v[j] - m);
        sum += v[j];
    }
    #pragma unroll
    for (int off = 16; off > 0; off >>= 1)
        sum += __shfl_xor(sum, off, 32);

    const float inv = 1.0f / sum;
    float* orow = p + (size_t)b * FDIM;
    #pragma unroll
    for (int j = 0; j < 16; ++j)
        orow[lane + 32 * j] = v[j] * inv;
}

// ---------------------------------------------------------------------------
// Kernel 2: per-batch reductions Sp = Sigma@p, q = Sigma^T@p, s = p^T Sigma p,
// computed on the matrix pipe with V_WMMA_F32_16X16X4_F32 (f32 in/out).
//
// For a 16x16 Sigma tile at (i0,k0):
//   Sp:  D += A(Sigma 16x4 chunk) x B(p broadcast across columns)
//        -> every column of D holds Sp partials for rows i0..i0+15
//   q:   D += A(p broadcast across rows) x B(Sigma 4x16 chunk)
//        -> every row of D holds q partials for cols k0..k0+15
// One block (128 threads = 4 waves) per batch; q accumulated in LDS (ds_add).
// ---------------------------------------------------------------------------
__global__ void rvs_reduce_kernel(const float* __restrict__ Sigma,
                                  const float* __restrict__ p,
                                  float* __restrict__ Sp,   // [B,F]
                                  float* __restrict__ q,    // [B,F]
                                  float* __restrict__ s) {  // [B]
    __shared__ float q_lds[FDIM];
    __shared__ float sp_lds[FDIM];
    __shared__ float red[4];

    const int b    = blockIdx.x;
    const int tid  = threadIdx.x;      // 0..127
    const int wave = tid >> 5;         // 0..3
    const int lane = tid & 31;
    const int half = lane >> 4;        // 0: lanes 0-15, 1: lanes 16-31
    const int l16  = lane & 15;

    const float* Sb = Sigma + (size_t)b * FDIM * FDIM;
    const float* pb = p + (size_t)b * FDIM;

    for (int i = tid; i < FDIM; i += 128) q_lds[i] = 0.0f;
    __syncthreads();

    // Each wave owns row-blocks rb = wave, wave+4, ... (uniform control flow,
    // EXEC all-ones at every WMMA as required).
    for (int rb = wave; rb < FDIM / 16; rb += 4) {
        const int i0 = rb * 16;

        // A fragments for q: A[j][i] = p[i0+4*ii+i] (broadcast down rows).
        v2f aq[4];
        #pragma unroll
        for (int ii = 0; ii < 4; ++ii) {
            const int base = i0 + 4 * ii + 2 * half;
            aq[ii].x = pb[base + 0];
            aq[ii].y = pb[base + 1];
        }

        v8f dsp = {};   // Sp accumulator for rows i0..i0+15, chained over all k
        for (int k0 = 0; k0 < FDIM; k0 += 16) {
            // ---- Sp: Sigma as A, broadcast-p as B, K chained 4x ----
            #pragma unroll
            for (int kk = 0; kk < 4; ++kk) {
                const int colA = k0 + 4 * kk + 2 * half;
                const int rowA = i0 + l16;
                v2f a, bm;
                a.x  = Sb[(size_t)rowA * FDIM + colA + 0];
                a.y  = Sb[(size_t)rowA * FDIM + colA + 1];
                bm.x = pb[colA + 0];
                bm.y = pb[colA + 1];
                dsp = __builtin_amdgcn_wmma_f32_16x16x4_f32(
                        false, a, false, bm, (short)0, dsp, false, false);
            }
            // ---- q: broadcast-p as A, Sigma as B, K chained 4x over i ----
            v8f dq = {};
            #pragma unroll
            for (int ii = 0; ii < 4; ++ii) {
                const int rowB = i0 + 4 * ii + 2 * half;
                const int colB = k0 + l16;
                v2f bm;
                bm.x = Sb[(size_t)(rowB + 0) * FDIM + colB];
                bm.y = Sb[(size_t)(rowB + 1) * FDIM + colB];
                dq = __builtin_amdgcn_wmma_f32_16x16x4_f32(
                        false, aq[ii], false, bm, (short)0, dq, false, false);
            }
            // D row 0 (VGPR 0, lanes 0-15) holds q partials for cols k0+l16.
            if (half == 0)
                atomicAdd(&q_lds[k0 + l16], dq[0]);   // ds_add_f32
        }
        // All columns of dsp are identical: lane 0 holds rows 0-7 (VGPR r),
        // lane 16 holds rows 8-15.
        if (lane == 0) {
            #pragma unroll
            for (int r = 0; r < 8; ++r) sp_lds[i0 + r] = dsp[r];
        } else if (lane == 16) {
            #pragma unroll
            for (int r = 0; r < 8; ++r) sp_lds[i0 + 8 + r] = dsp[r];
        }
    }
    __syncthreads();

    // Emit Sp/q, and reduce s = p^T Sp across the block.
    float acc = 0.0f;
    for (int i = tid; i < FDIM; i += 128) {
        const float spv = sp_lds[i];
        Sp[(size_t)b * FDIM + i] = spv;
        q [(size_t)b * FDIM + i] = q_lds[i];
        acc += pb[i] * spv;
    }
    #pragma unroll
    for (int off = 16; off > 0; off >>= 1)
        acc += __shfl_xor(acc, off, 32);
    if (lane == 0) red[wave] = acc;
    __syncthreads();
    if (tid == 0) s[b] = red[0] + red[1] + red[2] + red[3];
}

// ---------------------------------------------------------------------------
// Kernel 3: streaming transform (bandwidth-bound pass).
//   Sigma_out[i,k] = p_i * p_k * (Sigma[i,k] - Sp_i - q_k + s)
// One block (512 threads) per batch; p/q/Sp cached in LDS; fully coalesced
// 128-bit loads, prefetch ahead, non-temporal 128-bit stores (write-once data).
// ---------------------------------------------------------------------------
__global__ void rvs_finalize_kernel(const float* __restrict__ Sigma,
                                    const float* __restrict__ p,
                                    const float* __restrict__ Sp,
                                    const float* __restrict__ q,
                                    const float* __restrict__ s,
                                    float* __restrict__ out) {
    __shared__ float p_lds[FDIM];
    __shared__ float q_lds[FDIM];
    __shared__ float sp_lds[FDIM];

    const int b   = blockIdx.x;
    const int tid = threadIdx.x;               // 0..511
    p_lds[tid]  = p [(size_t)b * FDIM + tid];
    q_lds[tid]  = q [(size_t)b * FDIM + tid];
    sp_lds[tid] = Sp[(size_t)b * FDIM + tid];
    __syncthreads();

    const float sv   = s[b];
    const size_t base = (size_t)b * FDIM * FDIM;

    const int k4     = (tid * 4) & (FDIM - 1);   // first column of this lane's quad
    const int rowOff = (tid * 4) >> 9;           // 0..3: row within 4-row group

    const v4f pk = *(const v4f*)&p_lds[k4];      // ds_load_b128
    const v4f qk = *(const v4f*)&q_lds[k4];

    for (int j = 0; j < FDIM / 4; ++j) {         // 4 rows per iteration
        const int row   = j * 4 + rowOff;
        const size_t off = base + (size_t)row * FDIM + k4;

        __builtin_prefetch(Sigma + off + 4 * FDIM, 0, 1);  // global_prefetch

        const v4f   sig = *(const v4f*)(Sigma + off);      // global_load_b128
        const float pi  = p_lds[row];
        const float c   = sv - sp_lds[row];

        v4f t;
        t.x = pi * pk.x * (sig.x + c - qk.x);
        t.y = pi * pk.y * (sig.y + c - qk.y);
        t.z = pi * pk.z * (sig.z + c - qk.z);
        t.w = pi * pk.w * (sig.w + c - qk.w);

        __builtin_nontemporal_store(t, (v4f*)(out + off)); // NT b128 store
    }
}

// ---------------------------------------------------------------------------
// Launcher. d_in[0]=mu [B,F] f32, d_in[1]=Sigma [B,F,F] f32.
// d_out = [ p (B*F) | Sigma_out (B*F*F) ] f32.
// Workspace: Sp (B*F) | q (B*F) | s (B)  floats  (~1.05 MB).
// ---------------------------------------------------------------------------
extern "C" void kernel_launch(void* const* d_in, const int* in_sizes, int n_in,
                              void* d_out, int out_size, void* d_ws, size_t ws_size,
                              hipStream_t stream) {
    const float* mu    = (const float*)d_in[0];
    const float* Sigma = (const float*)d_in[1];
    float* out = (float*)d_out;

    float* p    = out;                                 // [B,F]
    float* Sout = out + (size_t)BATCH * FDIM;          // [B,F,F]

    float* ws = (float*)d_ws;
    float* Sp = ws;                                    // [B,F]
    float* q  = ws + (size_t)BATCH * FDIM;             // [B,F]
    float* s  = ws + (size_t)2 * BATCH * FDIM;         // [B]

    rvs_softmax_kernel <<<BATCH,  32, 0, stream>>>(mu, p);
    rvs_reduce_kernel  <<<BATCH, 128, 0, stream>>>(Sigma, p, Sp, q, s);
    rvs_finalize_kernel<<<BATCH, 512, 0, stream>>>(Sigma, p, Sp, q, s, Sout);
}